// DirectVolumeRenderer_71468255805791
// MI455X (gfx1250) — compile-verified
//
#include <hip/hip_runtime.h>
#include <math.h>

typedef __attribute__((ext_vector_type(2))) float v2f;
typedef __attribute__((ext_vector_type(8))) float v8f;
// 8-byte vector load with only 4-byte alignment guarantee (x-pairs of voxels)
typedef float f2u __attribute__((ext_vector_type(2), aligned(4)));

#define IMG_N   256       // IMG_W == IMG_H == 256
#define N_PTS   320
#define VOL_N   256
#define FOCAL   2.0f
#define MIN_D   2.0f
#define MAX_D   6.0f
#define EPS_EA  1e-10f
#define EPS_F   1e-8f
#define DENSITY 0.1f

// half = vox*(N-1)*0.5, vox = 3/256  ->  invHalf = 1/1.494140625
#define INV_HALF (512.0f / 765.0f)

// ---------------------------------------------------------------------------
// One ray-march step: trilinear sample (4x global_load_b64 on the interior
// fast path) + transmittance update.
// ---------------------------------------------------------------------------
__device__ __forceinline__ void dvr_step(
    const float* __restrict__ vol, int i, float dstep,
    float dirx, float diry, float dirz,
    float offx, float offy, float offz,
    float& Tacc, float& acc)
{
    const float d  = fmaf((float)i, dstep, MIN_D);
    const float lx = fmaf(d, dirx, offx);
    const float ly = fmaf(d, diry, offy);
    const float lz = fmaf(d, dirz, offz);

    // grid_sample coords, align_corners=True: (l+1)*0.5*(N-1)
    const float ix = fmaf(lx, 127.5f, 127.5f);
    const float iy = fmaf(ly, 127.5f, 127.5f);
    const float iz = fmaf(lz, 127.5f, 127.5f);

    const float fx0 = floorf(ix), fy0 = floorf(iy), fz0 = floorf(iz);
    const float fx = ix - fx0, fy = iy - fy0, fz = iz - fz0;
    const int x0 = (int)fx0, y0 = (int)fy0, z0 = (int)fz0;

    const float wx1 = fx, wx0 = 1.0f - fx;
    const float wy1 = fy, wy0 = 1.0f - fy;
    const float wz1 = fz, wz0 = 1.0f - fz;

    float feat, wsum;
    if (((unsigned)x0 < 255u) & ((unsigned)y0 < 255u) & ((unsigned)z0 < 255u)) {
        // fast interior path: all 8 corners valid, occupancy = 1.
        // 4 x-adjacent pairs -> 4 dword-aligned 8-byte loads (global_load_b64).
        const float* p = vol + ((z0 * VOL_N + y0) * VOL_N + x0);
        const f2u v00 = *(const f2u*)(p);
        const f2u v01 = *(const f2u*)(p + VOL_N);
        const f2u v10 = *(const f2u*)(p + VOL_N * VOL_N);
        const f2u v11 = *(const f2u*)(p + VOL_N * VOL_N + VOL_N);
        feat = wz0 * (wy0 * (wx0 * v00.x + wx1 * v00.y) +
                      wy1 * (wx0 * v01.x + wx1 * v01.y)) +
               wz1 * (wy0 * (wx0 * v10.x + wx1 * v10.y) +
                      wy1 * (wx0 * v11.x + wx1 * v11.y));
        wsum = 1.0f;
    } else {
        // boundary path: zero padding + per-corner validity (occupancy)
        feat = 0.0f; wsum = 0.0f;
        const int x1 = x0 + 1, y1 = y0 + 1, z1 = z0 + 1;
        const bool vx0 = (unsigned)x0 < (unsigned)VOL_N;
        const bool vx1 = (unsigned)x1 < (unsigned)VOL_N;
        const bool vy0 = (unsigned)y0 < (unsigned)VOL_N;
        const bool vy1 = (unsigned)y1 < (unsigned)VOL_N;
        const bool vz0 = (unsigned)z0 < (unsigned)VOL_N;
        const bool vz1 = (unsigned)z1 < (unsigned)VOL_N;
        const int xc0 = x0 < 0 ? 0 : (x0 > 255 ? 255 : x0);
        const int xc1 = x1 < 0 ? 0 : (x1 > 255 ? 255 : x1);
        const int yc0 = y0 < 0 ? 0 : (y0 > 255 ? 255 : y0);
        const int yc1 = y1 < 0 ? 0 : (y1 > 255 ? 255 : y1);
        const int zc0 = z0 < 0 ? 0 : (z0 > 255 ? 255 : z0);
        const int zc1 = z1 < 0 ? 0 : (z1 > 255 ? 255 : z1);
        float w;
        w = wz0 * wy0 * wx0; if (vz0 & vy0 & vx0) { wsum += w; feat += w * vol[(zc0 * VOL_N + yc0) * VOL_N + xc0]; }
        w = wz0 * wy0 * wx1; if (vz0 & vy0 & vx1) { wsum += w; feat += w * vol[(zc0 * VOL_N + yc0) * VOL_N + xc1]; }
        w = wz0 * wy1 * wx0; if (vz0 & vy1 & vx0) { wsum += w; feat += w * vol[(zc0 * VOL_N + yc1) * VOL_N + xc0]; }
        w = wz0 * wy1 * wx1; if (vz0 & vy1 & vx1) { wsum += w; feat += w * vol[(zc0 * VOL_N + yc1) * VOL_N + xc1]; }
        w = wz1 * wy0 * wx0; if (vz1 & vy0 & vx0) { wsum += w; feat += w * vol[(zc1 * VOL_N + yc0) * VOL_N + xc0]; }
        w = wz1 * wy0 * wx1; if (vz1 & vy0 & vx1) { wsum += w; feat += w * vol[(zc1 * VOL_N + yc0) * VOL_N + xc1]; }
        w = wz1 * wy1 * wx0; if (vz1 & vy1 & vx0) { wsum += w; feat += w * vol[(zc1 * VOL_N + yc1) * VOL_N + xc0]; }
        w = wz1 * wy1 * wx1; if (vz1 & vy1 & vx1) { wsum += w; feat += w * vol[(zc1 * VOL_N + yc1) * VOL_N + xc1]; }
    }

    const float sigma = DENSITY * wsum;
    acc  = fmaf(sigma * Tacc, feat, acc);       // weights[i] * feat[i]
    Tacc = Tacc * (1.0f + EPS_EA - sigma);      // running exclusive cumprod
}

// ---------------------------------------------------------------------------
// Kernel 1: ray march. One thread per pixel; ray directions computed with
// V_WMMA_F32_16X16X4_F32 (16 pixels per WMMA, fp32-exact), redistributed
// through LDS to the owning lanes.
// ---------------------------------------------------------------------------
__global__ __launch_bounds__(256) void dvr_render(
    const float* __restrict__ vol,   // 256^3
    const float* __restrict__ Rm,    // 3x3 row-major
    const float* __restrict__ Tv,    // 3
    float* __restrict__ rgb)         // 65536, [h*256 + w]
{
    __shared__ float sdir[256 * 4];

    const int tid  = threadIdx.x;
    const int lane = tid & 31;
    const int wv   = tid >> 5;
    const int blockBase = blockIdx.x * 256;

    // ---- B operand: B[j][n] = R[n][j]/half (n<3, j<3), else 0 -------------
    // 4x16 f32 B layout: VGPR0 = rows K=0 (lanes 0-15) / K=2 (16-31),
    //                    VGPR1 = rows K=1 (lanes 0-15) / K=3 (16-31).
    const int n  = lane & 15;
    const int kb = (lane < 16) ? 0 : 2;
    v2f b;
    b.x = (n < 3) ? Rm[n * 3 + kb] * INV_HALF : 0.0f;            // j = kb (<3)
    b.y = (n < 3 && kb == 0) ? Rm[n * 3 + 1] * INV_HALF : 0.0f;  // j = 1 or 3(=0)

#pragma unroll
    for (int g = 0; g < 2; ++g) {
        // A operand: A[m][j] = (xg/F, yg/F, 1, 0) for pixel m of this group.
        // Layout: lanes 0-15 hold (K=0,K=1); lanes 16-31 hold (K=2,K=3).
        const int pm = blockBase + wv * 32 + g * 16 + (lane & 15);
        const int ph = pm >> 8;
        const int pw = pm & 255;
        const float xg = fmaf((float)pw, 2.0f / 255.0f, -1.0f);
        const float yg = fmaf((float)ph, 2.0f / 255.0f, -1.0f);
        v2f a;
        if (lane < 16) { a.x = xg * (1.0f / FOCAL); a.y = yg * (1.0f / FOCAL); }
        else           { a.x = 1.0f;                a.y = 0.0f; }

        v8f c = {};
        c = __builtin_amdgcn_wmma_f32_16x16x4_f32(
                false, a, false, b, (short)0, c, false, false);

        // D[m][n]: VGPR r holds row m=r (lanes 0-15) / m=r+8 (lanes 16-31).
        if (n < 3) {
            const int mhi = (lane >= 16) ? 8 : 0;
#pragma unroll
            for (int r = 0; r < 8; ++r)
                sdir[(wv * 32 + g * 16 + mhi + r) * 4 + n] = c[r];
        }
    }
    __syncthreads();

    const float dirx = sdir[tid * 4 + 0];
    const float diry = sdir[tid * 4 + 1];
    const float dirz = sdir[tid * 4 + 2];

    // off = -(R/half) * T  (shared by all rays; compiler scalarizes to s_fmac)
    const float T0 = Tv[0], T1 = Tv[1], T2 = Tv[2];
    const float offx = -(T0 * Rm[0] + T1 * Rm[1] + T2 * Rm[2]) * INV_HALF;
    const float offy = -(T0 * Rm[3] + T1 * Rm[4] + T2 * Rm[5]) * INV_HALF;
    const float offz = -(T0 * Rm[6] + T1 * Rm[7] + T2 * Rm[8]) * INV_HALF;

    float Tacc = 1.0f;
    float acc  = 0.0f;
    const float dstep = (MAX_D - MIN_D) / (float)(N_PTS - 1);

    // 2-way unrolled so two samples' gathers (8x b64) are in flight per trip;
    // early termination is numerically inert: once Tacc < 1e-10 every
    // remaining per-step contribution is < 1e-11 (sigma<=0.1, feat<=1).
    for (int i = 0; i < N_PTS; i += 2) {
        dvr_step(vol, i,     dstep, dirx, diry, dirz, offx, offy, offz, Tacc, acc);
        dvr_step(vol, i + 1, dstep, dirx, diry, dirz, offx, offy, offz, Tacc, acc);
        if (Tacc < 1e-10f) break;
    }

    rgb[blockBase + tid] = acc;
}

// ---------------------------------------------------------------------------
// Kernel 2: per-block reduction (sum, sumsq, min, max) — deterministic tree.
// ---------------------------------------------------------------------------
__global__ __launch_bounds__(256) void dvr_reduce1(
    const float* __restrict__ rgb, float* __restrict__ part)
{
    __shared__ float ssum[256], ssq[256], smin[256], smax[256];
    const int tid = threadIdx.x;
    const float x = rgb[blockIdx.x * 256 + tid];
    ssum[tid] = x; ssq[tid] = x * x; smin[tid] = x; smax[tid] = x;
    __syncthreads();
    for (int s = 128; s > 0; s >>= 1) {
        if (tid < s) {
            ssum[tid] += ssum[tid + s];
            ssq[tid]  += ssq[tid + s];
            smin[tid]  = fminf(smin[tid], smin[tid + s]);
            smax[tid]  = fmaxf(smax[tid], smax[tid + s]);
        }
        __syncthreads();
    }
    if (tid == 0) {
        part[blockIdx.x * 4 + 0] = ssum[0];
        part[blockIdx.x * 4 + 1] = ssq[0];
        part[blockIdx.x * 4 + 2] = smin[0];
        part[blockIdx.x * 4 + 3] = smax[0];
    }
}

// ---------------------------------------------------------------------------
// Kernel 3: final reduction + normalization constants.
// ---------------------------------------------------------------------------
__global__ __launch_bounds__(256) void dvr_reduce2(
    const float* __restrict__ part, float* __restrict__ fin)
{
    __shared__ float ssum[256], ssq[256], smin[256], smax[256];
    const int tid = threadIdx.x;
    ssum[tid] = part[tid * 4 + 0];
    ssq[tid]  = part[tid * 4 + 1];
    smin[tid] = part[tid * 4 + 2];
    smax[tid] = part[tid * 4 + 3];
    __syncthreads();
    for (int s = 128; s > 0; s >>= 1) {
        if (tid < s) {
            ssum[tid] += ssum[tid + s];
            ssq[tid]  += ssq[tid + s];
            smin[tid]  = fminf(smin[tid], smin[tid + s]);
            smax[tid]  = fmaxf(smax[tid], smax[tid + s]);
        }
        __syncthreads();
    }
    if (tid == 0) {
        const float N = (float)(IMG_N * IMG_N);
        const float sum = ssum[0], sq = ssq[0], mn = smin[0], mx = smax[0];
        const float mean = sum / N;
        float var = (sq - sum * sum / N) / (N - 1.0f);   // ddof=1
        var = fmaxf(var, 0.0f);
        const float inv = 1.0f / (sqrtf(var) + EPS_F);
        const float smn = (mn - mean) * inv;
        const float smx = (mx - mean) * inv;
        fin[0] = mean;
        fin[1] = inv;
        fin[2] = smn;
        fin[3] = 1.0f / (smx - smn + EPS_F);
    }
}

// ---------------------------------------------------------------------------
// Kernel 4: normalize + (0,2,1) transpose, coalesced writes.
// out[w*256 + h] = norm(rgb[h*256 + w])
// ---------------------------------------------------------------------------
__global__ __launch_bounds__(256) void dvr_finalize(
    const float* __restrict__ rgb, const float* __restrict__ fin,
    float* __restrict__ out)
{
    const int idx = blockIdx.x * 256 + threadIdx.x;   // output index = w*256 + h
    const int w = idx >> 8;
    const int h = idx & 255;
    const float mean = fin[0], inv = fin[1], smn = fin[2], dinv = fin[3];
    const float s = (rgb[h * IMG_N + w] - mean) * inv;
    out[idx] = (s - smn + EPS_F) * dinv;
}

// ---------------------------------------------------------------------------
extern "C" void kernel_launch(void* const* d_in, const int* in_sizes, int n_in,
                              void* d_out, int out_size, void* d_ws, size_t ws_size,
                              hipStream_t stream)
{
    const float* vol = (const float*)d_in[0];   // (1,1,256,256,256) f32
    const float* Rm  = (const float*)d_in[1];   // (1,3,3) f32
    const float* Tv  = (const float*)d_in[2];   // (1,3) f32
    float* out = (float*)d_out;                 // 65536 f32

    float* rgb  = (float*)d_ws;                 // 65536 floats
    float* part = rgb + IMG_N * IMG_N;          // 256*4 floats
    float* fin  = part + 256 * 4;               // 4 floats

    dvr_render  <<<256, 256, 0, stream>>>(vol, Rm, Tv, rgb);
    dvr_reduce1 <<<256, 256, 0, stream>>>(rgb, part);
    dvr_reduce2 <<<1,   256, 0, stream>>>(part, fin);
    dvr_finalize<<<256, 256, 0, stream>>>(rgb, fin, out);
}